// C4PureTransformer_62380105007321
// MI455X (gfx1250) — compile-verified
//
#include <hip/hip_runtime.h>
#include <hip/hip_bf16.h>
#include <stdint.h>

// ---------------------------------------------------------------------------
// One step of a C4-style VM.
// Outputs (concatenated flat, float): [pc, sp, bp, ax, memory[2^25], halted]
// Cost model: ~0 FLOPs, 128MB read + 128MB write => pure HBM streaming
// (~11us @ 23.3 TB/s). CDNA5 path used: async global->LDS data mover
// (global_load_async_to_lds_b128 / ASYNCcnt / s_wait_asynccnt), with
// non-temporal hints since the 256MB stream has zero reuse (> 192MB L2).
// ---------------------------------------------------------------------------

#define MEM_ELEMS 33554432LL  // 2^25 byte-cells

typedef float __attribute__((ext_vector_type(4))) f32x4;  // native vectors for
typedef int   __attribute__((ext_vector_type(4))) i32x4;  // NT store builtin

// ------------------------- bulk copy/convert kernel -------------------------
constexpr int TPB         = 256;              // 8 waves (wave32)
constexpr int VEC         = 4;                // int32 cells per 16B async chunk
constexpr int CHUNKS      = 4;                // outstanding async chunks / thread
constexpr int BLOCK_ELEMS = TPB * VEC * CHUNKS;  // 4096 cells / block

__global__ void __launch_bounds__(TPB)
vm_copy_convert(const int* __restrict__ in, float* __restrict__ out) {
  __shared__ int lds[BLOCK_ELEMS];  // 16 KB staging (well under 320KB/WGP)

  const int tid  = threadIdx.x;
  const int base = blockIdx.x * BLOCK_ELEMS;
  const uint32_t lds_base = (uint32_t)(uintptr_t)(&lds[0]);  // low 32b = LDS offset

  // Issue CHUNKS outstanding 16B/lane async transfers: global -> LDS.
  // Tracked on ASYNCcnt; lanes only consume their own data, so a per-wave
  // s_wait_asynccnt is sufficient (no barrier needed). NT hint: stream-once
  // data, do not retain in L2.
#pragma unroll
  for (int j = 0; j < CHUNKS; ++j) {
    const int idx          = (j * TPB + tid) * VEC;           // coalesced per j
    const int* gptr        = in + base + idx;
    const uint32_t laddr   = lds_base + (uint32_t)(idx * (int)sizeof(int));
    asm volatile("global_load_async_to_lds_b128 %0, %1, off th:TH_LOAD_NT"
                 :
                 : "v"(laddr), "v"(gptr)
                 : "memory");
  }
  asm volatile("s_wait_asynccnt 0" ::: "memory");

  // Drain LDS -> convert int32 byte-cell to float -> coalesced 16B NT stores.
#pragma unroll
  for (int j = 0; j < CHUNKS; ++j) {
    const int idx = (j * TPB + tid) * VEC;
    i32x4 v = *(const i32x4*)&lds[idx];
    f32x4 f;
    f.x = (float)v.x; f.y = (float)v.y; f.z = (float)v.z; f.w = (float)v.w;
    __builtin_nontemporal_store(f, (f32x4*)(out + base + idx));
  }
}

// ----------------------------- VM step helpers ------------------------------
__device__ __forceinline__ long long vm_clamp(long long a) {
  return a < 0 ? 0 : (a > (MEM_ELEMS - 1) ? (MEM_ELEMS - 1) : a);
}

__device__ __forceinline__ long long vm_read8(const int* m, long long addr) {
  unsigned long long v = 0;
#pragma unroll
  for (int k = 0; k < 8; ++k) {
    unsigned b = (unsigned)m[vm_clamp(addr + k)] & 0xFFu;
    v |= (unsigned long long)b << (8 * k);
  }
  return (long long)v;
}

__device__ __forceinline__ void vm_write8(float* outmem, long long addr, long long value) {
#pragma unroll
  for (int k = 0; k < 8; ++k) {
    unsigned b = (unsigned)(((unsigned long long)value >> (8 * k)) & 0xFFull);
    outmem[vm_clamp(addr + k)] = (float)b;
  }
}

// Python-style floor division / modulo on int64 (jnp semantics).
__device__ __forceinline__ long long vm_fdiv(long long a, long long b) {
  if (b == -1) return (long long)(0ULL - (unsigned long long)a);  // avoid UB on INT64_MIN
  long long q = a / b, r = a % b;
  if (r != 0 && ((r < 0) != (b < 0))) --q;
  return q;
}
__device__ __forceinline__ long long vm_fmod(long long a, long long b) {
  if (b == -1) return 0;
  long long r = a % b;
  if (r != 0 && ((r < 0) != (b < 0))) r += b;
  return r;
}
__device__ __forceinline__ long long wadd(long long a, long long b) {
  return (long long)((unsigned long long)a + (unsigned long long)b);
}
__device__ __forceinline__ long long wsub(long long a, long long b) {
  return (long long)((unsigned long long)a - (unsigned long long)b);
}
__device__ __forceinline__ long long wmul(long long a, long long b) {
  return (long long)((unsigned long long)a * (unsigned long long)b);
}

// ------------------------------ VM step kernel ------------------------------
__global__ void vm_step(const int* __restrict__ pcp, const int* __restrict__ spp,
                        const int* __restrict__ bpp, const int* __restrict__ axp,
                        const int* __restrict__ mem, float* __restrict__ out) {
  if (threadIdx.x != 0 || blockIdx.x != 0) return;

  const long long pc = (long long)pcp[0];
  const long long sp = (long long)spp[0];
  const long long bp = (long long)bpp[0];
  const long long ax = (long long)axp[0];

  // fetch / decode
  const long long instruction = vm_read8(mem, pc);
  const long long opcode      = instruction & 255;           // 0..255
  const long long imm         = instruction >> 8;            // arithmetic shift

  const long long stack_top = vm_read8(mem, sp);
  const long long mem_at_ax = vm_read8(mem, ax);
  const long long ret_pc    = vm_read8(mem, bp + 8);
  const long long bp_stack  = vm_read8(mem, bp);
  const long long pc8       = wadd(pc, 8);
  const long long ax_safe   = (ax == 0) ? 1 : ax;

  const int op = (int)(opcode > 38 ? 38 : opcode);           // clipped dispatch index

  // new_pc
  long long new_pc;
  switch (op) {
    case 2: case 3: new_pc = imm; break;
    case 4:  new_pc = (ax == 0) ? imm : pc8; break;
    case 5:  new_pc = (ax != 0) ? imm : pc8; break;
    case 8:  new_pc = ret_pc; break;
    case 38: new_pc = pc; break;
    default: new_pc = pc8; break;
  }

  // new_sp
  long long new_sp;
  switch (op) {
    case 3: case 13: new_sp = wsub(sp, 8); break;
    case 6:  new_sp = wsub(wsub(sp, 8), imm); break;
    case 7:  new_sp = wadd(sp, imm); break;
    case 8:  new_sp = wadd(bp, 16); break;
    case 11: case 12: new_sp = wadd(sp, 8); break;
    default:
      new_sp = (op >= 14 && op <= 29) ? wadd(sp, 8) : sp;
      break;
  }

  // new_bp
  long long new_bp = bp;
  if (op == 6) new_bp = wsub(sp, 8);
  else if (op == 8) new_bp = bp_stack;

  // new_ax
  long long new_ax;
  switch (op) {
    case 0:  new_ax = wadd(bp, imm); break;              // LEA
    case 1:  new_ax = imm; break;                        // IMM
    case 9:  new_ax = mem_at_ax; break;                  // LI
    case 10: new_ax = mem_at_ax & 255; break;            // LC
    case 14: new_ax = wadd(stack_top, ax); break;
    case 15: new_ax = wsub(stack_top, ax); break;
    case 16: new_ax = wmul(stack_top, ax); break;
    case 17: new_ax = vm_fdiv(stack_top, ax_safe); break;
    case 18: new_ax = vm_fmod(stack_top, ax_safe); break;
    case 19: new_ax = stack_top | ax; break;
    case 20: new_ax = stack_top ^ ax; break;
    case 21: new_ax = stack_top & ax; break;
    case 22: new_ax = (long long)((unsigned long long)stack_top << (int)(ax & 63)); break;
    case 23: new_ax = stack_top >> (int)(ax & 63); break; // arithmetic
    case 24: new_ax = (stack_top == ax) ? 1 : 0; break;
    case 25: new_ax = (stack_top != ax) ? 1 : 0; break;
    case 26: new_ax = (stack_top <  ax) ? 1 : 0; break;
    case 27: new_ax = (stack_top >  ax) ? 1 : 0; break;
    case 28: new_ax = (stack_top <= ax) ? 1 : 0; break;
    case 29: new_ax = (stack_top >= ax) ? 1 : 0; break;
    default:
      new_ax = (op >= 30 && op <= 37) ? 0 : ax;          // syscalls -> 0; else keep
      break;
  }

  // masked memory writes on top of copied image (raw opcode, like reference)
  float* outmem = out + 4;
  const int opr    = (int)opcode;
  const bool is_psh = (opr == 13), is_jsr = (opr == 3), is_ent = (opr == 6);
  const bool is_si  = (opr == 11), is_sc  = (opr == 12);

  if (is_psh || is_jsr || is_ent) {
    const long long push_value = is_psh ? ax : (is_jsr ? pc8 : bp);
    vm_write8(outmem, wsub(sp, 8), push_value);
  }
  if (is_si) vm_write8(outmem, stack_top, ax);
  if (is_sc) outmem[vm_clamp(stack_top)] = (float)(unsigned)(ax & 255);

  // registers + halted
  out[0] = (float)new_pc;
  out[1] = (float)new_sp;
  out[2] = (float)new_bp;
  out[3] = (float)new_ax;
  out[4 + MEM_ELEMS] = (opcode == 38) ? 1.0f : 0.0f;
}

// --------------------------------- launcher ---------------------------------
extern "C" void kernel_launch(void* const* d_in, const int* in_sizes, int n_in,
                              void* d_out, int out_size, void* d_ws, size_t ws_size,
                              hipStream_t stream) {
  (void)in_sizes; (void)n_in; (void)out_size; (void)d_ws; (void)ws_size;

  const int* pc  = (const int*)d_in[0];
  const int* sp  = (const int*)d_in[1];
  const int* bp  = (const int*)d_in[2];
  const int* ax  = (const int*)d_in[3];
  const int* mem = (const int*)d_in[4];
  float* out = (float*)d_out;

  // 1) stream the 256MB memory image: async global->LDS -> cvt -> NT global
  const int blocks = (int)(MEM_ELEMS / BLOCK_ELEMS);  // 8192, exact cover
  vm_copy_convert<<<blocks, TPB, 0, stream>>>(mem, out + 4);

  // 2) single-lane interpreter step patches registers + <=17 bytes + halted
  vm_step<<<1, 32, 0, stream>>>(pc, sp, bp, ax, mem, out);
}